// LengthRegulator_74732430950600
// MI455X (gfx1250) — compile-verified
//
#include <hip/hip_runtime.h>
#include <hip/hip_bf16.h>

// ---------------------------------------------------------------------------
// LengthRegulator for MI455X (gfx1250).
// Pure data-movement problem: 352 MB of output stores dominate; ~17 us floor
// at 23.3 TB/s HBM. Strategy:
//   K1: per-batch LDS scan of duration -> csum, mel_len (to d_out tail + ws)
//   K2: direct scatter of token index (duration <= 7) -> idx16[b, f]
//   K3: per (batch, 8-channel tile) block: TDM tensor_load_to_lds stages the
//       32KB x tile in LDS (wave0 issues, s_wait_tensorcnt fences) while all
//       waves stage the 14KB idx tile with b128 loads; then coalesced
//       non-temporal float4 stores of the expanded frames.
// ---------------------------------------------------------------------------

#define B_DIM 32
#define C_DIM 384
#define T_DIM 1024
#define F_DIM 7168
#define CT 8                      // channel rows per block in gather kernel
#define OUT_ELEMS (B_DIM * C_DIM * F_DIM)

typedef float          v4f  __attribute__((ext_vector_type(4)));
typedef unsigned int   v4u  __attribute__((ext_vector_type(4)));
typedef int            v4i  __attribute__((ext_vector_type(4)));
typedef int            v8i  __attribute__((ext_vector_type(8)));

#if defined(__has_builtin)
#if __has_builtin(__builtin_amdgcn_tensor_load_to_lds)
#define HAVE_TDM 1
#endif
#endif

// ---------------------------------------------------------------------------
// Kernel 3 (placed first so the disasm snippet shows it): expand + gather.
// One block per (batch, 8-row channel tile).
// ---------------------------------------------------------------------------
__global__ __launch_bounds__(256) void lr_gather_kernel(
    const float* __restrict__ x, const int* __restrict__ mel,
    const unsigned short* __restrict__ idx, float* __restrict__ out) {
  __shared__ float xs[CT * T_DIM];          // 32 KB: 8 contiguous x rows
  __shared__ unsigned short is[F_DIM];      // 14 KB: frame -> token index

  const int tid = threadIdx.x;
  const int b  = blockIdx.x / (C_DIM / CT);
  const int cg = blockIdx.x % (C_DIM / CT);
  const size_t xbase = ((size_t)b * C_DIM + (size_t)cg * CT) * T_DIM;

#ifdef HAVE_TDM
  // Wave 0 issues one TDM DMA of the full 32KB contiguous x tile into LDS.
  if (tid < 32) {
    unsigned long long ga =
        (unsigned long long)(const void*)(x + xbase);         // byte address
    unsigned int lds_off =
        (unsigned int)(unsigned long long)(void*)&xs[0];      // low 32 bits = LDS offset
    // D# group 0: count=1 | lds_addr | global_addr[56:0] | type=2
    v4u g0;
    g0.x = 1u;                                                // count=1, user mode
    g0.y = lds_off;
    g0.z = (unsigned int)(ga & 0xffffffffull);
    g0.w = (unsigned int)((ga >> 32) & 0x01ffffffull) | 0x80000000u; // type=2
    // D# group 1: data_size=4B, 1-D tensor: dim0=8192, stride0=8192,
    // tile_dim0=8192, tile_dim1=1 (dim1 fields set to benign values).
    v8i g1;
    g1[0] = 0x00020000;            // workgroup_mask=0, data_size=2 (4 bytes)
    g1[1] = (int)(8192u << 16);    // tensor_dim0[15:0] in bits[63:48]
    g1[2] = 0x00010000;            // tensor_dim0 hi = 0, tensor_dim1 = 1
    g1[3] = (int)(8192u << 16);    // tile_dim0 = 8192 in bits[127:112]
    g1[4] = 1;                     // tile_dim1 = 1, tile_dim2 = 0
    g1[5] = 8192;                  // tensor_dim0_stride low 32
    g1[6] = (int)(8192u << 16);    // stride0 hi = 0, tensor_dim1_stride lo16
    g1[7] = 0;                     // tensor_dim1_stride hi
    v4i z4 = {0, 0, 0, 0};
    v8i z8 = {0, 0, 0, 0, 0, 0, 0, 0};
    __builtin_amdgcn_tensor_load_to_lds(g0, g1, z4, z4, z8, 0);
  }
#else
  // Fallback: cooperative b128 copy of the x tile into LDS.
  {
    const v4f* src = (const v4f*)(x + xbase);
    v4f* dst = (v4f*)&xs[0];
    for (int i = tid; i < (CT * T_DIM) / 4; i += 256) dst[i] = src[i];
  }
#endif

  // All waves stage the idx tile (7168 u16 = 896 x 16B) with b128 loads,
  // overlapping the TDM DMA. Both base and row stride are 16B-aligned.
  {
    const v4u* src16 = (const v4u*)(idx + (size_t)b * F_DIM);
    v4u* dst16 = (v4u*)&is[0];
    for (int i = tid; i < (F_DIM * 2) / 16; i += 256) dst16[i] = src16[i];
  }

#ifdef HAVE_TDM
  if (tid < 32) __builtin_amdgcn_s_wait_tensorcnt(0);
#endif
  __syncthreads();

  const int melb = mel[b];
  float* outbase = out + ((size_t)b * C_DIM + (size_t)cg * CT) * F_DIM;

  // 7168/4 = 1792 float4 columns; 256 threads -> 7 iterations.
  const unsigned long long* is64 = (const unsigned long long*)&is[0];
  for (int i = tid; i < F_DIM / 4; i += 256) {
    const int f0 = i << 2;
    // One ds_load_b64 fetches all 4 u16 indices (8B-aligned: f0 % 4 == 0).
    const unsigned long long packed = is64[i];
    // Clamp (frames >= mel_len hold unwritten/poison idx) and mask.
    int j0 = (int)(packed       & 0xffff); j0 = (j0 > T_DIM - 1) ? (T_DIM - 1) : j0;
    int j1 = (int)((packed >> 16) & 0xffff); j1 = (j1 > T_DIM - 1) ? (T_DIM - 1) : j1;
    int j2 = (int)((packed >> 32) & 0xffff); j2 = (j2 > T_DIM - 1) ? (T_DIM - 1) : j2;
    int j3 = (int)((packed >> 48) & 0xffff); j3 = (j3 > T_DIM - 1) ? (T_DIM - 1) : j3;
    const bool m0 = (f0 + 0) < melb;
    const bool m1 = (f0 + 1) < melb;
    const bool m2 = (f0 + 2) < melb;
    const bool m3 = (f0 + 3) < melb;
#pragma unroll
    for (int r = 0; r < CT; ++r) {
      const float* xr = &xs[r * T_DIM];
      v4f v;
      v.x = m0 ? xr[j0] : 0.0f;
      v.y = m1 ? xr[j1] : 0.0f;
      v.z = m2 ? xr[j2] : 0.0f;
      v.w = m3 ? xr[j3] : 0.0f;
      // Streaming (non-temporal) b128 store: 352MB write stream, keep L2 clean.
      __builtin_nontemporal_store(
          v, (v4f*)(outbase + (size_t)r * F_DIM + f0));
    }
  }
}

// ---------------------------------------------------------------------------
// Kernel 1: inclusive scan of duration[b, 0:1024] (Hillis-Steele in LDS).
// ---------------------------------------------------------------------------
__global__ __launch_bounds__(T_DIM) void lr_scan_kernel(
    const int* __restrict__ dur, int* __restrict__ csum,
    int* __restrict__ mel, int* __restrict__ out_tail) {
  __shared__ int s[T_DIM];
  const int b = blockIdx.x;
  const int t = threadIdx.x;
  s[t] = dur[(size_t)b * T_DIM + t];
  __syncthreads();
#pragma unroll
  for (int off = 1; off < T_DIM; off <<= 1) {
    int v = (t >= off) ? s[t - off] : 0;
    __syncthreads();
    s[t] += v;
    __syncthreads();
  }
  csum[(size_t)b * T_DIM + t] = s[t];
  if (t == T_DIM - 1) {
    mel[b] = s[t];
    out_tail[b] = s[t];   // mel_len as raw int32 bits at the tail of d_out
  }
}

// ---------------------------------------------------------------------------
// Kernel 2: scatter token id into idx16. Token t covers frames
// [csum[t-1], csum[t]) -- at most 7 frames (duration in [0,8)).
// Frames >= mel_len are never written: gather kernel clamps + masks them.
// ---------------------------------------------------------------------------
__global__ __launch_bounds__(T_DIM) void lr_expand_kernel(
    const int* __restrict__ csum, unsigned short* __restrict__ idx) {
  const int b = blockIdx.x;
  const int t = threadIdx.x;
  const int* cs = csum + (size_t)b * T_DIM;
  int end = cs[t];
  int start = (t == 0) ? 0 : cs[t - 1];
  if (end > F_DIM) end = F_DIM;
  unsigned short* row = idx + (size_t)b * F_DIM;
  for (int f = start; f < end; ++f) row[f] = (unsigned short)t;
}

// ---------------------------------------------------------------------------
// Launch
// ---------------------------------------------------------------------------
extern "C" void kernel_launch(void* const* d_in, const int* in_sizes, int n_in,
                              void* d_out, int out_size, void* d_ws, size_t ws_size,
                              hipStream_t stream) {
  const float* x   = (const float*)d_in[0];          // (32, 384, 1024) f32
  const int*   dur = (const int*)d_in[1];            // (32, 1024) i32
  // d_in[2] = max_len scalar (device ptr); value is 7168 per reference setup.

  float* out = (float*)d_out;                        // (32,384,7168) f32 + 32 i32

  // Workspace layout: csum[32*1024] i32 | mel[32] i32 | idx[32*7168] u16
  int* csum = (int*)d_ws;
  int* mel  = csum + (size_t)B_DIM * T_DIM;
  unsigned short* idx = (unsigned short*)(mel + B_DIM);

  lr_scan_kernel<<<B_DIM, T_DIM, 0, stream>>>(dur, csum, mel,
                                              (int*)d_out + OUT_ELEMS);
  lr_expand_kernel<<<B_DIM, T_DIM, 0, stream>>>(csum, idx);
  lr_gather_kernel<<<B_DIM * (C_DIM / CT), 256, 0, stream>>>(x, mel, idx, out);
}